// GCN_15925738734178
// MI455X (gfx1250) — compile-verified
//
#include <hip/hip_runtime.h>
#include <hip/hip_bf16.h>

typedef __attribute__((ext_vector_type(2))) float v2f;
typedef __attribute__((ext_vector_type(8))) float v8f;

#define N_NODES 80000
#define N_EDGES 1280000

// ---------------------------------------------------------------- utilities

__global__ __launch_bounds__(256) void gcn_zero_f4(float4* __restrict__ p, int n4) {
    int i = blockIdx.x * 256 + threadIdx.x;
    if (i < n4) p[i] = make_float4(0.f, 0.f, 0.f, 0.f);
}

__global__ __launch_bounds__(256) void gcn_degree(const int* __restrict__ src,
                                                  float* __restrict__ deg, int E) {
    int e = blockIdx.x * 256 + threadIdx.x;
    if (e < E) atomicAdd(&deg[src[e]], 1.0f);
}

__global__ __launch_bounds__(256) void gcn_norm_finish(float* __restrict__ d, int n) {
    int i = blockIdx.x * 256 + threadIdx.x;
    if (i < n) {
        float v = d[i];
        v = v < 1.0f ? 1.0f : v;
        d[i] = rsqrtf(v);
    }
}

// ------------------------------------------------- edge scatter (gather+add)
// One wave32 per edge; lane moves 4 consecutive floats of the 128-wide row.
// Fuses the *first* norm scaling (norm[src]) into the gather.
__global__ __launch_bounds__(256)
void gcn_scatter(const float* __restrict__ h, const float* __restrict__ norm,
                 const int* __restrict__ src, const int* __restrict__ dst,
                 float* __restrict__ agg, int E) {
    int tid = blockIdx.x * 256 + threadIdx.x;
    int e = tid >> 5;
    if (e >= E) return;
    int lane = tid & 31;
    int s = src[e];
    int d = dst[e];
    float ns = norm[s];
    float4 v = *((const float4*)(h + (size_t)s * 128) + lane);
    float* ap = agg + (size_t)d * 128 + lane * 4;
    atomicAdd(ap + 0, v.x * ns);
    atomicAdd(ap + 1, v.y * ns);
    atomicAdd(ap + 2, v.z * ns);
    atomicAdd(ap + 3, v.w * ns);
}

// ------------------------------------------------- fused (norm*A) @ W + b GEMM
// fp32 WMMA 16x16x4. Each block computes a 16-row slab; wave w owns output
// columns [16w, 16w+16). A tile staged in LDS (stride 132 -> bank-disjoint
// half-wave v2f reads); B panel (128 x 16) preloaded into 32 v2f registers.
//
// A frag (16x4 f32): lane m=lane&15 holds row m; VGPR0/1 = K = 2*hi + {0,1},
//   hi = lane>>4 selects K-halves {0,1} vs {2,3} of the 4-wide K step.
// B frag (4x16 f32): lane n=lane&15 holds col n; VGPR0/1 = rows K = 2*hi+{0,1}.
// C/D (16x16 f32): VGPR r -> row M = r + 8*hi, col N = lane&15.
template <int NCOLS, bool SCALE, bool RELU>
__global__ __launch_bounds__(NCOLS * 32)
void gcn_gemm_wmma(const float* __restrict__ A, const float* __restrict__ norm,
                   const float* __restrict__ W, const float* __restrict__ bias,
                   float* __restrict__ out) {
    constexpr int OUTC = NCOLS * 16;  // 128 (layers) or 64 (final)
    constexpr int LDA = 132;          // padded LDS row stride (floats)
    __shared__ float As[16 * LDA];

    const int tid = threadIdx.x;
    const int lane = tid & 31;
    const int wave = tid >> 5;
    const int hi = lane >> 4;   // 0 for lanes 0-15, 1 for lanes 16-31
    const int lo = lane & 15;
    const int rowBase = blockIdx.x * 16;  // N = 80000 is a multiple of 16

    // Cooperative load of the 16x128 A tile (optionally scaled by norm[row]).
    for (int i = tid; i < 16 * 32; i += NCOLS * 32) {
        int row = i >> 5;
        int c4 = (i & 31) << 2;
        float4 v = *(const float4*)(A + (size_t)(rowBase + row) * 128 + c4);
        float s = SCALE ? norm[rowBase + row] : 1.0f;
        float* d = &As[row * LDA + c4];
        d[0] = v.x * s;
        d[1] = v.y * s;
        d[2] = v.z * s;
        d[3] = v.w * s;
    }

    // Preload this wave's 128x16 B panel into registers (L2-hot weight).
    const int col = wave * 16 + lo;
    v2f bReg[32];
#pragma unroll
    for (int kt = 0; kt < 32; ++kt) {
        int k0 = 4 * kt + 2 * hi;
        v2f b;
        b.x = W[(size_t)k0 * OUTC + col];
        b.y = W[(size_t)(k0 + 1) * OUTC + col];
        bReg[kt] = b;
    }

    // Accumulator starts at the bias (bias depends only on the output column).
    v8f c;
    float bv = bias[col];
#pragma unroll
    for (int r = 0; r < 8; ++r) c[r] = bv;

    __syncthreads();

#pragma unroll
    for (int kt = 0; kt < 32; ++kt) {
        v2f a = *(const v2f*)&As[lo * LDA + 4 * kt + 2 * hi];
        c = __builtin_amdgcn_wmma_f32_16x16x4_f32(
            /*neg_a=*/false, a, /*neg_b=*/false, bReg[kt],
            /*c_mod=*/(short)0, c, /*reuse_a=*/false, /*reuse_b=*/false);
    }

#pragma unroll
    for (int r = 0; r < 8; ++r) {
        float v = c[r];
        if (RELU) v = v > 0.f ? v : 0.f;
        out[(size_t)(rowBase + r + 8 * hi) * OUTC + col] = v;
    }
}

// ---------------------------------------------------------------- launcher

extern "C" void kernel_launch(void* const* d_in, const int* in_sizes, int n_in,
                              void* d_out, int out_size, void* d_ws, size_t ws_size,
                              hipStream_t stream) {
    const float* feat = (const float*)d_in[0];
    const float* W1   = (const float*)d_in[1];
    const float* b1   = (const float*)d_in[2];
    const float* W2   = (const float*)d_in[3];
    const float* b2   = (const float*)d_in[4];
    const float* W3   = (const float*)d_in[5];
    const float* b3   = (const float*)d_in[6];
    const int*   src  = (const int*)d_in[7];
    const int*   dst  = (const int*)d_in[8];
    float* out = (float*)d_out;

    const int N = N_NODES;
    const int E = N_EDGES;

    // Workspace carve-up: norm (N f32) | agg (N*128 f32) | h (N*128 f32)
    float* norm = (float*)d_ws;
    float* agg  = norm + N;                 // byte offset 320000 (16B aligned)
    float* hbuf = agg + (size_t)N * 128;

    const int NB_NORM = (N / 4 + 255) / 256;       // zero N floats as float4
    const int NB_AGG  = (N * 128 / 4) / 256;       // 10000 blocks
    const int NB_EDGE = (E + 255) / 256;
    const int NB_SCAT = (E * 32) / 256;            // one wave per edge
    const int NB_ROW  = N / 16;                    // 5000 row tiles

    // degrees -> norm = rsqrt(max(deg,1))
    gcn_zero_f4<<<NB_NORM, 256, 0, stream>>>((float4*)norm, N / 4);
    gcn_degree<<<NB_EDGE, 256, 0, stream>>>(src, norm, E);
    gcn_norm_finish<<<(N + 255) / 256, 256, 0, stream>>>(norm, N);

    // Layer 1: agg = scatter(norm*feat); h = relu((norm*agg) @ W1 + b1)
    gcn_zero_f4<<<NB_AGG, 256, 0, stream>>>((float4*)agg, N * 128 / 4);
    gcn_scatter<<<NB_SCAT, 256, 0, stream>>>(feat, norm, src, dst, agg, E);
    gcn_gemm_wmma<8, true, true><<<NB_ROW, 256, 0, stream>>>(agg, norm, W1, b1, hbuf);

    // Layer 2: agg = scatter(norm*h); h = relu((norm*agg) @ W2 + b2)
    gcn_zero_f4<<<NB_AGG, 256, 0, stream>>>((float4*)agg, N * 128 / 4);
    gcn_scatter<<<NB_SCAT, 256, 0, stream>>>(hbuf, norm, src, dst, agg, E);
    gcn_gemm_wmma<8, true, true><<<NB_ROW, 256, 0, stream>>>(agg, norm, W2, b2, hbuf);

    // Output head: out = h @ W3 + b3   (128 -> 64, 4 waves / 64 cols)
    gcn_gemm_wmma<4, false, false><<<NB_ROW, 128, 0, stream>>>(hbuf, norm, W3, b3, out);
}